// GlobalFeatureExtractor_80161269612823
// MI455X (gfx1250) — compile-verified
//
#include <hip/hip_runtime.h>
#include <math.h>

#define B_ROWS 16384
#define IN_F   1024
#define DET    4
#define H1_F   64
#define H2_F   32
#define N1     (DET*H1_F)   /* 256 */
#define N2     (DET*H2_F)   /* 128 */
#define EPS_BN 1e-5f

typedef __attribute__((ext_vector_type(2))) float v2f;
typedef __attribute__((ext_vector_type(4))) float v4f;
typedef __attribute__((ext_vector_type(8))) float v8f;

__device__ __forceinline__ v8f wmma_f32(v2f a, v2f b, v8f c) {
  // D = A(16x4,f32) * B(4x16,f32) + C(16x16,f32)
  return __builtin_amdgcn_wmma_f32_16x16x4_f32(false, a, false, b, (short)0, c, false, false);
}

__device__ __forceinline__ float sigmoidf_(float t) {
  return 1.0f / (1.0f + __expf(-t));
}

// ---------------------------------------------------------------------------
// Column statistics, pass 1: per-row-group partial sum / sum-of-squares.
// Deterministic (no float atomics). src is row-major [rows x cols].
// ---------------------------------------------------------------------------
__global__ __launch_bounds__(256) void partial_colstats(
    const float* __restrict__ src, int rows, int cols, int rpg,
    float* __restrict__ psum, float* __restrict__ psq) {
  int col = blockIdx.x * 256 + threadIdx.x;
  int g = blockIdx.y;
  if (col >= cols) return;
  int r0 = g * rpg;
  int r1 = min(rows, r0 + rpg);
  float s = 0.f, q = 0.f;
  for (int r = r0; r < r1; ++r) {
    float v = src[(size_t)r * cols + col];
    s += v;
    q += v * v;
  }
  psum[(size_t)g * cols + col] = s;
  psq[(size_t)g * cols + col] = q;
}

// Pass 2: reduce G partials -> mean, rstd per column (biased var, eps=1e-5).
__global__ __launch_bounds__(256) void finalize_stats(
    const float* __restrict__ psum, const float* __restrict__ psq,
    int cols, int G, float inv_rows,
    float* __restrict__ meanv, float* __restrict__ rstdv) {
  int col = blockIdx.x * 256 + threadIdx.x;
  if (col >= cols) return;
  float s = 0.f, q = 0.f;
  for (int g = 0; g < G; ++g) {
    s += psum[(size_t)g * cols + col];
    q += psq[(size_t)g * cols + col];
  }
  float m = s * inv_rows;
  float var = q * inv_rows - m * m;
  meanv[col] = m;
  rstdv[col] = rsqrtf(var + EPS_BN);
}

// sc[c] = rstd[c]*gamma[c];  sh[c] = beta[c] - mean[c]*sc[c]
__global__ __launch_bounds__(256) void make_affine(
    const float* __restrict__ meanv, const float* __restrict__ rstdv,
    const float* __restrict__ gamma, const float* __restrict__ beta,
    int cols, float* __restrict__ sc, float* __restrict__ sh) {
  int c = blockIdx.x * 256 + threadIdx.x;
  if (c >= cols) return;
  float s = rstdv[c] * gamma[c];
  sc[c] = s;
  sh[c] = beta[c] - meanv[c] * s;
}

// ---------------------------------------------------------------------------
// Fold bn0 (tiled-x batchnorm) into W1:
//   W1p[n][k] = W1[d][o][k] * rstd0[k] * g0[d*IN+k]          (n = d*64+o)
//   bias1[n]  = sum_k W1[d][o][k] * (be0[d*IN+k] - mean0[k]*rstd0[k]*g0[d*IN+k])
// ---------------------------------------------------------------------------
__global__ __launch_bounds__(256) void prep_w1(
    const float* __restrict__ W1, const float* __restrict__ g0,
    const float* __restrict__ be0, const float* __restrict__ mean0,
    const float* __restrict__ rstd0,
    float* __restrict__ W1p, float* __restrict__ bias1) {
  int n = blockIdx.x;                 // 0..255
  int d = n / H1_F, o = n % H1_F;
  const float* wrow = W1 + ((size_t)d * H1_F + o) * IN_F;
  float acc = 0.f;
  for (int k = threadIdx.x; k < IN_F; k += 256) {
    float w = wrow[k];
    float s = rstd0[k] * g0[d * IN_F + k];
    W1p[(size_t)n * IN_F + k] = w * s;
    acc += w * (be0[d * IN_F + k] - mean0[k] * s);
  }
  __shared__ float red[256];
  red[threadIdx.x] = acc;
  __syncthreads();
  for (int s = 128; s > 0; s >>= 1) {
    if (threadIdx.x < (unsigned)s) red[threadIdx.x] += red[threadIdx.x + s];
    __syncthreads();
  }
  if (threadIdx.x == 0) bias1[n] = red[0];
}

// ---------------------------------------------------------------------------
// GEMM1: Y1[16384 x 256] = X[16384 x 1024] * W1p^T + bias1
// Block: 256 thr (8 waves), tile 128(M) x 64(N), K staged 32 wide in LDS.
// Wave grid 4(M) x 2(N); each wave holds 2x2 f32 16x16 accumulators.
// LDS row stride 34: v2f fragments 8B-aligned & bank-conflict-free.
// ---------------------------------------------------------------------------
#define LDA1 34
__global__ __launch_bounds__(256) void gemm1(
    const float* __restrict__ X, const float* __restrict__ W1p,
    const float* __restrict__ bias1, float* __restrict__ Y1) {
  __shared__ float As[128 * LDA1];
  __shared__ float Bs[64 * LDA1];
  const int t = threadIdx.x;
  const int lane = t & 31, w = t >> 5;
  const int lane_r = lane & 15, lane_h = lane >> 4;
  const int m0 = blockIdx.x * 128;
  const int n0 = blockIdx.y * 64;
  const int wm = (w & 3) * 32;   // wave M offset
  const int wn = (w >> 2) * 32;  // wave N offset

  v8f c[2][2] = {{{}, {}}, {{}, {}}};

  for (int k0 = 0; k0 < IN_F; k0 += 32) {
    // stage A: 128x32 floats (1024 float4, 4/thread)
#pragma unroll
    for (int i = 0; i < 4; ++i) {
      int q = t + i * 256;
      int row = q >> 3, c4 = (q & 7) * 4;
      v4f v = *(const v4f*)&X[(size_t)(m0 + row) * IN_F + k0 + c4];
      int base = row * LDA1 + c4;
      As[base + 0] = v.x; As[base + 1] = v.y;
      As[base + 2] = v.z; As[base + 3] = v.w;
    }
    // stage B: 64x32 floats (512 float4, 2/thread)
#pragma unroll
    for (int i = 0; i < 2; ++i) {
      int q = t + i * 256;
      int row = q >> 3, c4 = (q & 7) * 4;
      v4f v = *(const v4f*)&W1p[(size_t)(n0 + row) * IN_F + k0 + c4];
      int base = row * LDA1 + c4;
      Bs[base + 0] = v.x; Bs[base + 1] = v.y;
      Bs[base + 2] = v.z; Bs[base + 3] = v.w;
    }
    __syncthreads();
#pragma unroll
    for (int ks = 0; ks < 32; ks += 4) {
      int kk = ks + 2 * lane_h;  // even -> 8B aligned
      v2f a0 = *(const v2f*)&As[(wm + lane_r) * LDA1 + kk];
      v2f a1 = *(const v2f*)&As[(wm + 16 + lane_r) * LDA1 + kk];
      v2f b0 = *(const v2f*)&Bs[(wn + lane_r) * LDA1 + kk];
      v2f b1 = *(const v2f*)&Bs[(wn + 16 + lane_r) * LDA1 + kk];
      c[0][0] = wmma_f32(a0, b0, c[0][0]);
      c[0][1] = wmma_f32(a0, b1, c[0][1]);
      c[1][0] = wmma_f32(a1, b0, c[1][0]);
      c[1][1] = wmma_f32(a1, b1, c[1][1]);
    }
    __syncthreads();
  }
  // epilogue: C layout per ISA: M = v + 8*lane_h, N = lane_r
#pragma unroll
  for (int im = 0; im < 2; ++im) {
#pragma unroll
    for (int in = 0; in < 2; ++in) {
      int col = n0 + wn + in * 16 + lane_r;
      float bv = bias1[col];
#pragma unroll
      for (int v = 0; v < 8; ++v) {
        int row = m0 + wm + im * 16 + v + 8 * lane_h;
        Y1[(size_t)row * N1 + col] = c[im][in][v] + bv;
      }
    }
  }
}

// ---------------------------------------------------------------------------
// GEMM2 (block-diagonal): per detector d,
//   out[b, d*32+o] = sum_j sigmoid(bn1(Y1[b, d*64+j])) * W2[d][o][j]
// bn1+sigmoid applied while staging A. Block: 128(M) x 32(N), K=64 single
// stage. 8 waves along M; 2 accumulators (N halves) per wave.
// ---------------------------------------------------------------------------
#define LDA2 66
__global__ __launch_bounds__(256) void gemm2(
    const float* __restrict__ Y1, const float* __restrict__ W2,
    const float* __restrict__ sc1, const float* __restrict__ sh1,
    float* __restrict__ out) {
  __shared__ float As[128 * LDA2];
  __shared__ float Bs[32 * LDA2];
  const int t = threadIdx.x;
  const int lane = t & 31, w = t >> 5;  // 8 waves along M
  const int lane_r = lane & 15, lane_h = lane >> 4;
  const int m0 = blockIdx.x * 128;
  const int d = blockIdx.y;

  // stage A with bn1 affine + sigmoid: 128x64 floats (2048 float4, 8/thread)
#pragma unroll
  for (int i = 0; i < 8; ++i) {
    int q = t + i * 256;
    int row = q >> 4, c4 = (q & 15) * 4;
    int gcol = d * H1_F + c4;
    v4f v = *(const v4f*)&Y1[(size_t)(m0 + row) * N1 + gcol];
    int base = row * LDA2 + c4;
    As[base + 0] = sigmoidf_(v.x * sc1[gcol + 0] + sh1[gcol + 0]);
    As[base + 1] = sigmoidf_(v.y * sc1[gcol + 1] + sh1[gcol + 1]);
    As[base + 2] = sigmoidf_(v.z * sc1[gcol + 2] + sh1[gcol + 2]);
    As[base + 3] = sigmoidf_(v.w * sc1[gcol + 3] + sh1[gcol + 3]);
  }
  // stage B: W2[d] 32x64 (512 float4, 2/thread)
#pragma unroll
  for (int i = 0; i < 2; ++i) {
    int q = t + i * 256;
    int row = q >> 4, c4 = (q & 15) * 4;
    v4f v = *(const v4f*)&W2[((size_t)d * H2_F + row) * H1_F + c4];
    int base = row * LDA2 + c4;
    Bs[base + 0] = v.x; Bs[base + 1] = v.y;
    Bs[base + 2] = v.z; Bs[base + 3] = v.w;
  }
  __syncthreads();

  v8f c0 = {}, c1 = {};
#pragma unroll
  for (int ks = 0; ks < 64; ks += 4) {
    int kk = ks + 2 * lane_h;
    v2f a = *(const v2f*)&As[(w * 16 + lane_r) * LDA2 + kk];
    v2f b0 = *(const v2f*)&Bs[lane_r * LDA2 + kk];
    v2f b1 = *(const v2f*)&Bs[(16 + lane_r) * LDA2 + kk];
    c0 = wmma_f32(a, b0, c0);
    c1 = wmma_f32(a, b1, c1);
  }
#pragma unroll
  for (int in = 0; in < 2; ++in) {
    v8f cc = in ? c1 : c0;
    int col = d * H2_F + in * 16 + lane_r;
#pragma unroll
    for (int v = 0; v < 8; ++v) {
      int row = m0 + w * 16 + v + 8 * lane_h;
      out[(size_t)row * N2 + col] = cc[v];
    }
  }
}

// Final: out = sigmoid(out*sc2[col] + sh2[col]) in place.
__global__ __launch_bounds__(256) void bn_sigmoid_inplace(
    float* __restrict__ out, const float* __restrict__ sc,
    const float* __restrict__ sh, int cols, int n) {
  int i = blockIdx.x * 256 + threadIdx.x;
  if (i >= n) return;
  int c = i % cols;
  out[i] = sigmoidf_(out[i] * sc[c] + sh[c]);
}

// ---------------------------------------------------------------------------
extern "C" void kernel_launch(void* const* d_in, const int* in_sizes, int n_in,
                              void* d_out, int out_size, void* d_ws, size_t ws_size,
                              hipStream_t stream) {
  const float* x   = (const float*)d_in[0];
  const float* g0  = (const float*)d_in[1];
  const float* be0 = (const float*)d_in[2];
  const float* W1  = (const float*)d_in[3];
  const float* g1  = (const float*)d_in[4];
  const float* be1 = (const float*)d_in[5];
  const float* W2  = (const float*)d_in[6];
  const float* g2  = (const float*)d_in[7];
  const float* be2 = (const float*)d_in[8];
  float* out = (float*)d_out;

  const int G = 64;                       // row groups for stats
  const int RPG = B_ROWS / G;             // 256

  float* ws = (float*)d_ws;
  float* psum  = ws;                      // 64*1024
  float* psq   = psum + G * IN_F;         // 64*1024
  float* mean0 = psq + G * IN_F;          // 1024
  float* rstd0 = mean0 + IN_F;            // 1024
  float* mean1 = rstd0 + IN_F;            // 256
  float* rstd1 = mean1 + N1;              // 256
  float* sc1   = rstd1 + N1;              // 256
  float* sh1   = sc1 + N1;                // 256
  float* mean2 = sh1 + N1;                // 128
  float* rstd2 = mean2 + N2;              // 128
  float* sc2   = rstd2 + N2;              // 128
  float* sh2   = sc2 + N2;                // 128
  float* bias1 = sh2 + N2;                // 256
  float* W1p   = bias1 + N1;              // 256*1024
  float* Y1    = W1p + (size_t)N1 * IN_F; // 16384*256

  const float invB = 1.0f / (float)B_ROWS;

  // bn0 stats on x columns (tiled-x stats == x stats per column)
  partial_colstats<<<dim3(IN_F / 256, G), 256, 0, stream>>>(x, B_ROWS, IN_F, RPG, psum, psq);
  finalize_stats<<<IN_F / 256, 256, 0, stream>>>(psum, psq, IN_F, G, invB, mean0, rstd0);

  // fold bn0 into W1
  prep_w1<<<N1, 256, 0, stream>>>(W1, g0, be0, mean0, rstd0, W1p, bias1);

  // GEMM1: Y1 = X * W1p^T + bias1
  gemm1<<<dim3(B_ROWS / 128, N1 / 64), 256, 0, stream>>>(x, W1p, bias1, Y1);

  // bn1 stats + affine
  partial_colstats<<<dim3(1, G), 256, 0, stream>>>(Y1, B_ROWS, N1, RPG, psum, psq);
  finalize_stats<<<1, 256, 0, stream>>>(psum, psq, N1, G, invB, mean1, rstd1);
  make_affine<<<1, 256, 0, stream>>>(mean1, rstd1, g1, be1, N1, sc1, sh1);

  // GEMM2 (bn1+sigmoid fused on A load) -> pre-bn2 activations in d_out
  gemm2<<<dim3(B_ROWS / 128, DET), 256, 0, stream>>>(Y1, W2, sc1, sh1, out);

  // bn2 stats + affine, then in-place bn2+sigmoid
  partial_colstats<<<dim3(1, G), 256, 0, stream>>>(out, B_ROWS, N2, RPG, psum, psq);
  finalize_stats<<<1, 256, 0, stream>>>(psum, psq, N2, G, invB, mean2, rstd2);
  make_affine<<<1, 256, 0, stream>>>(mean2, rstd2, g2, be2, N2, sc2, sh2);

  int n_out = B_ROWS * N2;
  bn_sigmoid_inplace<<<n_out / 256, 256, 0, stream>>>(out, sc2, sh2, N2, n_out);
}